// ActorNetwork_30812095382098
// MI455X (gfx1250) — compile-verified
//
#include <hip/hip_runtime.h>
#include <hip/hip_bf16.h>
#include <math.h>

typedef __attribute__((ext_vector_type(2))) float v2f;
typedef __attribute__((ext_vector_type(8))) float v8f;

#define GRAPH_SIZE 40
#define N_HOSTS 13
#define ACTION_SPACE 145
#define IN_DIM 128
#define H1DIM 256
#define H2DIM 64
#define APAD 4          // LDS row padding (floats): stride%64==4 -> conflict-free
#define KMAX 256

static inline int cdiv_i(long long a, long long b) { return (int)((a + b - 1) / b); }

// ---------------------------------------------------------------------------
// degree / dinv
// ---------------------------------------------------------------------------
__global__ void k_deg_init(float* deg, int N) {
  int i = blockIdx.x * blockDim.x + threadIdx.x;
  if (i < N) deg[i] = 1.0f;  // self loop
}

__global__ void k_deg_edges(const int* __restrict__ dst, float* deg, int E) {
  int e = blockIdx.x * blockDim.x + threadIdx.x;
  if (e < E) {
    __hip_atomic_fetch_add(&deg[dst[e]], 1.0f, __ATOMIC_RELAXED,
                           __HIP_MEMORY_SCOPE_AGENT);
  }
}

__global__ void k_dinv(float* deg, int N) {
  int i = blockIdx.x * blockDim.x + threadIdx.x;
  if (i < N) deg[i] = 1.0f / sqrtf(deg[i]);  // deg >= 1 always (self loop)
}

// ---------------------------------------------------------------------------
// fp32 WMMA GEMM: C[M x Nc] = A[M x K] * W[K x Nc], row major, fp32.
// One block = Nc/64 waves; the 16 x K A tile is staged once into LDS with
// coalesced b128 loads and shared by all waves (each wave owns a 16x64 column
// supertile, 4 accumulators).  A fragments come from LDS via b64 reads with a
// K+4 row stride (bank = 4*r (+kk), conflict-free across the wave).
// Requires M%16==0, Nc%64==0, K%4==0, K<=KMAX.
// ---------------------------------------------------------------------------
__global__ __launch_bounds__(128) void k_gemm_wmma_f32(
    const float* __restrict__ A, const float* __restrict__ W,
    float* __restrict__ C, int K, int Nc) {
  __shared__ float Al[16 * (KMAX + APAD)];

  const int tid = threadIdx.x;
  const int lane = tid & 31;
  const int wave = tid >> 5;
  const int m0 = blockIdx.x * 16;
  const int n0 = wave * 64;
  const int r = lane & 15;            // A row within tile / B-C column
  const int kk = (lane >> 4) << 1;    // lanes 16-31 carry K=2,3 (ISA 7.12.2)
  const int lk = K + APAD;

  // Stage A tile (16 x K) into LDS, float4-wide, fully coalesced.
  const int k4 = K >> 2;
  for (int idx = tid; idx < 16 * k4; idx += blockDim.x) {
    const int row = idx / k4;
    const int c4 = idx - row * k4;
    const float4 v = ((const float4*)(A + (size_t)(m0 + row) * K))[c4];
    *(float4*)(&Al[row * lk + (c4 << 2)]) = v;
  }
  __syncthreads();

  v8f acc0 = {}, acc1 = {}, acc2 = {}, acc3 = {};
  const float* Arow = &Al[r * lk];

  for (int k0 = 0; k0 < K; k0 += 4) {
    const float2 ap = *(const float2*)(Arow + k0 + kk);  // ds_load b64
    v2f a;
    a.x = ap.x;
    a.y = ap.y;
    const float* Wr0 = W + (size_t)(k0 + kk) * Nc + n0 + r;
    const float* Wr1 = Wr0 + Nc;

    v2f b0; b0.x = Wr0[0];  b0.y = Wr1[0];
    v2f b1; b1.x = Wr0[16]; b1.y = Wr1[16];
    v2f b2; b2.x = Wr0[32]; b2.y = Wr1[32];
    v2f b3; b3.x = Wr0[48]; b3.y = Wr1[48];

    acc0 = __builtin_amdgcn_wmma_f32_16x16x4_f32(false, a, false, b0, (short)0,
                                                 acc0, false, false);
    acc1 = __builtin_amdgcn_wmma_f32_16x16x4_f32(false, a, false, b1, (short)0,
                                                 acc1, false, false);
    acc2 = __builtin_amdgcn_wmma_f32_16x16x4_f32(false, a, false, b2, (short)0,
                                                 acc2, false, false);
    acc3 = __builtin_amdgcn_wmma_f32_16x16x4_f32(false, a, false, b3, (short)0,
                                                 acc3, false, false);
  }

  // C layout: VGPR i -> M = i (lanes 0-15) / M = i+8 (lanes 16-31), N = lane%16
  const int row = m0 + (lane >> 4) * 8;
  const int col = n0 + r;
#pragma unroll
  for (int i = 0; i < 8; ++i) {
    float* Crow = C + (size_t)(row + i) * Nc + col;
    Crow[0]  = acc0[i];
    Crow[16] = acc1[i];
    Crow[32] = acc2[i];
    Crow[48] = acc3[i];
  }
}

// ---------------------------------------------------------------------------
// Aggregation: out[d] = sum over incoming edges of coef * xw[s]  (+ self loop)
// ---------------------------------------------------------------------------
__global__ void k_agg_self4(const float4* __restrict__ xw,
                            const float* __restrict__ dinv,
                            float4* __restrict__ out, long long total4,
                            int F4) {
  long long i = (long long)blockIdx.x * blockDim.x + threadIdx.x;
  if (i >= total4) return;
  int node = (int)(i / F4);
  float c = dinv[node];
  c = c * c;                        // self-loop coefficient dinv^2
  float4 v = xw[i];                 // global_load_b128
  v.x *= c; v.y *= c; v.z *= c; v.w *= c;
  out[i] = v;                       // global_store_b128
}

// One block per edge (blockDim == F).  e is block-uniform so src/dst/dinv
// become scalar (s_load) fetches; lanes only issue the xw load + fp32 atomic.
__global__ void k_agg_edges(const float* __restrict__ xw,
                            const int* __restrict__ src,
                            const int* __restrict__ dst,
                            const float* __restrict__ dinv,
                            float* __restrict__ out) {
  const int e = blockIdx.x;             // uniform -> SMEM loads
  const int s = src[e];
  const int d = dst[e];
  const float coef = dinv[s] * dinv[d];
  const int f = threadIdx.x;
  __hip_atomic_fetch_add(&out[(size_t)d * blockDim.x + f],
                         coef * xw[(size_t)s * blockDim.x + f],
                         __ATOMIC_RELAXED, __HIP_MEMORY_SCOPE_AGENT);
}

__global__ void k_bias_relu4(float4* __restrict__ h,
                             const float4* __restrict__ b, long long total4,
                             int F4) {
  long long i = (long long)blockIdx.x * blockDim.x + threadIdx.x;
  if (i >= total4) return;
  float4 v = h[i];
  float4 bb = b[(int)(i % F4)];
  v.x = fmaxf(v.x + bb.x, 0.0f);
  v.y = fmaxf(v.y + bb.y, 0.0f);
  v.z = fmaxf(v.z + bb.z, 0.0f);
  v.w = fmaxf(v.w + bb.w, 0.0f);
  h[i] = v;
}

// ---------------------------------------------------------------------------
// Readout: logits[g, a] = sum_{j<13, f<64} h2[(g*40+j)*64+f] * Wout[(j*64+f)*145+a]
// Host gather fused into the A-fragment load (hosts are nodes g*40 + 0..12).
// M = NB, K = 832, Nc = 145 (padded to 160 with guards).
// ---------------------------------------------------------------------------
__global__ __launch_bounds__(32) void k_gemm_out(
    const float* __restrict__ h2, const float* __restrict__ Wout,
    const float* __restrict__ bout, float* __restrict__ logits) {
  const int lane = threadIdx.x & 31;
  const int g0 = blockIdx.x * 16;
  const int n0 = blockIdx.y * 16;
  const int r = lane & 15;
  const int kk = (lane >> 4) << 1;
  const int col = n0 + r;
  const bool colok = (col < ACTION_SPACE);

  v8f acc = {};
  const int g = g0 + r;
  const size_t gbase = (size_t)g * GRAPH_SIZE * H2DIM;

  for (int k0 = 0; k0 < N_HOSTS * H2DIM; k0 += 4) {
    const int ka = k0 + kk;  // ka even, (ka&63)<=62 -> ka,ka+1 contiguous
    const float2 ap =
        *(const float2*)(h2 + gbase + (size_t)(ka >> 6) * H2DIM + (ka & 63));
    v2f a;
    a.x = ap.x;
    a.y = ap.y;
    v2f b;
    b.x = colok ? Wout[(size_t)ka * ACTION_SPACE + col] : 0.0f;
    b.y = colok ? Wout[(size_t)(ka + 1) * ACTION_SPACE + col] : 0.0f;
    acc = __builtin_amdgcn_wmma_f32_16x16x4_f32(false, a, false, b, (short)0,
                                                acc, false, false);
  }

  const int row = g0 + (lane >> 4) * 8;
  if (colok) {
    const float bb = bout[col];
#pragma unroll
    for (int i = 0; i < 8; ++i)
      logits[(size_t)(row + i) * ACTION_SPACE + col] = acc[i] + bb;
  }
}

// ---------------------------------------------------------------------------
// Softmax over 145 logits per graph, in place (exact expf).
// ---------------------------------------------------------------------------
__global__ void k_softmax(float* __restrict__ out) {
  __shared__ float red[256];
  const int g = blockIdx.x;
  const int t = threadIdx.x;
  float* row = out + (size_t)g * ACTION_SPACE;

  float v = (t < ACTION_SPACE) ? row[t] : -INFINITY;
  red[t] = v;
  __syncthreads();
  for (int s = 128; s > 0; s >>= 1) {
    if (t < s) red[t] = fmaxf(red[t], red[t + s]);
    __syncthreads();
  }
  const float m = red[0];
  __syncthreads();

  float e = (t < ACTION_SPACE) ? expf(v - m) : 0.0f;
  red[t] = e;
  __syncthreads();
  for (int s = 128; s > 0; s >>= 1) {
    if (t < s) red[t] += red[t + s];
    __syncthreads();
  }
  const float inv = 1.0f / red[0];
  if (t < ACTION_SPACE) row[t] = e * inv;
}

// ---------------------------------------------------------------------------
// Launch
// ---------------------------------------------------------------------------
extern "C" void kernel_launch(void* const* d_in, const int* in_sizes, int n_in,
                              void* d_out, int out_size, void* d_ws,
                              size_t ws_size, hipStream_t stream) {
  const float* x    = (const float*)d_in[0];
  const int*   ei   = (const int*)d_in[1];
  const float* W1   = (const float*)d_in[2];
  const float* b1   = (const float*)d_in[3];
  const float* W2   = (const float*)d_in[4];
  const float* b2   = (const float*)d_in[5];
  const float* Wout = (const float*)d_in[6];
  const float* bout = (const float*)d_in[7];
  float* out = (float*)d_out;

  const int N  = in_sizes[0] / IN_DIM;   // 81920
  const int E  = in_sizes[1] / 2;        // 655360
  const int NB = N / GRAPH_SIZE;         // 2048
  const int* src = ei;
  const int* dst = ei + E;

  // Workspace layout (layer-2 buffers alias layer-1 buffers).
  char* ws = (char*)d_ws;
  float* deg = (float*)ws;                                 // N floats (-> dinv)
  size_t off = ((size_t)N * sizeof(float) + 511) & ~(size_t)511;
  float* xw1 = (float*)(ws + off);                         // N x 256
  float* h1  = xw1 + (size_t)N * H1DIM;                    // N x 256
  float* xw2 = xw1;                                        // alias: N x 64
  float* h2  = h1;                                         // alias: N x 64

  const int T = 256;

  // ---- degree / dinv ----
  k_deg_init<<<cdiv_i(N, T), T, 0, stream>>>(deg, N);
  k_deg_edges<<<cdiv_i(E, T), T, 0, stream>>>(dst, deg, E);
  k_dinv<<<cdiv_i(N, T), T, 0, stream>>>(deg, N);

  // ---- layer 1: xw1 = x @ W1 ; h1 = D^-1/2 (A+I) D^-1/2 xw1 ; relu(+b1) ----
  k_gemm_wmma_f32<<<N / 16, (H1DIM / 64) * 32, 0, stream>>>(x, W1, xw1, IN_DIM,
                                                            H1DIM);
  {
    long long tot4 = (long long)N * H1DIM / 4;
    k_agg_self4<<<cdiv_i(tot4, T), T, 0, stream>>>((const float4*)xw1, deg,
                                                   (float4*)h1, tot4, H1DIM / 4);
    k_agg_edges<<<E, H1DIM, 0, stream>>>(xw1, src, dst, deg, h1);
    k_bias_relu4<<<cdiv_i(tot4, T), T, 0, stream>>>((float4*)h1,
                                                    (const float4*)b1, tot4,
                                                    H1DIM / 4);
  }

  // ---- layer 2: xw2 = h1 @ W2 ; h2 = agg(xw2) ; relu(+b2) ----
  k_gemm_wmma_f32<<<N / 16, (H2DIM / 64) * 32, 0, stream>>>(h1, W2, xw2, H1DIM,
                                                            H2DIM);
  {
    long long tot4 = (long long)N * H2DIM / 4;
    k_agg_self4<<<cdiv_i(tot4, T), T, 0, stream>>>((const float4*)xw2, deg,
                                                   (float4*)h2, tot4, H2DIM / 4);
    k_agg_edges<<<E, H2DIM, 0, stream>>>(xw2, src, dst, deg, h2);
    k_bias_relu4<<<cdiv_i(tot4, T), T, 0, stream>>>((float4*)h2,
                                                    (const float4*)b2, tot4,
                                                    H2DIM / 4);
  }

  // ---- readout GEMM (host gather fused) + softmax ----
  k_gemm_out<<<dim3(NB / 16, (ACTION_SPACE + 15) / 16), 32, 0, stream>>>(
      h2, Wout, bout, out);
  k_softmax<<<NB, 256, 0, stream>>>(out);
}